// GCN_Weighted_56057913147713
// MI455X (gfx1250) — compile-verified
//
#include <hip/hip_runtime.h>
#include <hip/hip_bf16.h>

typedef __attribute__((ext_vector_type(2))) float v2f;
typedef __attribute__((ext_vector_type(8))) float v8f;

#define CH 128

// ---------- degree / normalization ----------
__global__ __launch_bounds__(256) void k_fill1(float* __restrict__ deg, int n) {
  int i = blockIdx.x * blockDim.x + threadIdx.x;
  if (i < n) deg[i] = 1.0f;  // self-loop weight
}

__global__ __launch_bounds__(256) void k_degacc(const int* __restrict__ dst,
                                                const float* __restrict__ w,
                                                float* __restrict__ deg, int e) {
  int i = blockIdx.x * blockDim.x + threadIdx.x;
  if (i < e) atomicAdd(&deg[dst[i]], w[i]);
}

__global__ __launch_bounds__(256) void k_dinv(float* __restrict__ deg, int n) {
  int i = blockIdx.x * blockDim.x + threadIdx.x;
  if (i < n) {
    float d = deg[i];
    deg[i] = (d > 0.0f) ? rsqrtf(fmaxf(d, 1e-12f)) : 0.0f;
  }
}

__global__ __launch_bounds__(256) void k_norm(const int* __restrict__ src,
                                              const int* __restrict__ dst,
                                              const float* __restrict__ w,
                                              const float* __restrict__ dinv,
                                              float* __restrict__ norm, int e) {
  int i = blockIdx.x * blockDim.x + threadIdx.x;
  if (i < e) norm[i] = dinv[src[i]] * w[i] * dinv[dst[i]];
}

// ---------- GEMM: H[n x 128] = X[n x 128] @ W[128 x 128] via f32 WMMA ----------
// One wave computes a 16-row x 128-col strip: 8 accumulators, K-loop of x4 WMMAs.
__global__ __launch_bounds__(256) void k_gemm(const float* __restrict__ X,
                                              const float* __restrict__ W,
                                              float* __restrict__ H, int nrows) {
  int wave = (blockIdx.x * blockDim.x + threadIdx.x) >> 5;
  int lane = threadIdx.x & 31;
  int row0 = wave * 16;
  if (row0 >= nrows) return;
  int m    = lane & 15;   // A row / B col / D col within tile
  int half = lane >> 4;   // K-half selector

  if (row0 + 16 <= nrows) {
    const float* xrow = X + (size_t)(row0 + m) * CH;
    v8f acc[8] = {};
    for (int k = 0; k < CH; k += 4) {
      v2f a;
      a.x = xrow[k + 2 * half + 0];
      a.y = xrow[k + 2 * half + 1];
      const float* w0 = W + (size_t)(k + 2 * half) * CH;  // row k+2h
#pragma unroll
      for (int c = 0; c < 8; ++c) {
        v2f b;
        b.x = w0[c * 16 + m];        // W[k+2h][c*16+n]
        b.y = w0[CH + c * 16 + m];   // W[k+2h+1][c*16+n]
        acc[c] = __builtin_amdgcn_wmma_f32_16x16x4_f32(
            false, a, false, b, (short)0, acc[c], false, false);
      }
    }
#pragma unroll
    for (int c = 0; c < 8; ++c) {
      float* hp = H + (size_t)row0 * CH + c * 16 + m;
#pragma unroll
      for (int r = 0; r < 8; ++r)
        hp[(size_t)(r + 8 * half) * CH] = acc[c][r];  // D: row r+8h, col n
    }
  } else {
    // scalar tail (unused when nrows % 16 == 0)
    for (int r = row0; r < nrows; ++r)
      for (int c = lane; c < CH; c += 32) {
        float a = 0.0f;
        for (int k = 0; k < CH; ++k)
          a = fmaf(X[(size_t)r * CH + k], W[(size_t)k * CH + c], a);
        H[(size_t)r * CH + c] = a;
      }
  }
}

// ---------- edge gather + scaled scatter-add (L2-resident atomics) ----------
__global__ __launch_bounds__(256) void k_scatter(const int* __restrict__ src,
                                                 const int* __restrict__ dst,
                                                 const float* __restrict__ norm,
                                                 const float* __restrict__ h,
                                                 float* __restrict__ out, int e) {
  int wave = (blockIdx.x * blockDim.x + threadIdx.x) >> 5;
  int lane = threadIdx.x & 31;
  if (wave >= e) return;
  int s = src[wave], d = dst[wave];
  float w = norm[wave];
  float4 v = ((const float4*)(h + (size_t)s * CH))[lane];  // 512B gather / edge
  float* po = out + (size_t)d * CH + lane * 4;
  atomicAdd(po + 0, v.x * w);
  atomicAdd(po + 1, v.y * w);
  atomicAdd(po + 2, v.z * w);
  atomicAdd(po + 3, v.w * w);
}

// ---------- epilogue: + self-loop (dinv^2 * h) + bias (+ relu) ----------
__global__ __launch_bounds__(256) void k_epilogue(const float* __restrict__ agg,
                                                  const float* __restrict__ h,
                                                  const float* __restrict__ dinv,
                                                  const float* __restrict__ bias,
                                                  float* __restrict__ outx,
                                                  long long total, int relu) {
  long long i = (long long)blockIdx.x * blockDim.x + threadIdx.x;
  if (i >= total) return;
  int node = (int)(i >> 7);
  int c = (int)(i & (CH - 1));
  float di = dinv[node];
  float v = agg[i] + h[i] * di * di + bias[c];
  if (relu) v = fmaxf(v, 0.0f);
  outx[i] = v;
}

static inline int nblk(long long x) { return (int)((x + 255) / 256); }

extern "C" void kernel_launch(void* const* d_in, const int* in_sizes, int n_in,
                              void* d_out, int out_size, void* d_ws, size_t ws_size,
                              hipStream_t stream) {
  const float* X  = (const float*)d_in[0];
  const int* eix  = (const int*)d_in[1];
  const float* ew = (const float*)d_in[2];
  const float* W1 = (const float*)d_in[3];
  const float* b1 = (const float*)d_in[4];
  const float* W2 = (const float*)d_in[5];
  const float* b2 = (const float*)d_in[6];
  const float* W3 = (const float*)d_in[7];
  const float* b3 = (const float*)d_in[8];

  const int N = in_sizes[0] / CH;
  const int E = in_sizes[2];
  const int* src = eix;
  const int* dst = eix + E;

  // workspace: dinv[N] | norm[E] | bufA[N*128]   (~58 MB); d_out doubles as H.
  size_t Npad = ((size_t)N + 255) & ~(size_t)255;
  size_t Epad = ((size_t)E + 255) & ~(size_t)255;
  float* dinv = (float*)d_ws;
  float* norm = dinv + Npad;
  float* bufA = norm + Epad;
  float* H    = (float*)d_out;

  const long long NC = (long long)N * CH;
  const int gemm_blocks = (((N + 15) / 16) + 7) / 8;  // 8 waves/block, 16 rows/wave
  const int scat_blocks = (E + 7) / 8;                // 1 wave/edge

  // gcn_norm: deg -> dinv -> per-edge norm
  k_fill1 <<<nblk(N), 256, 0, stream>>>(dinv, N);
  k_degacc<<<nblk(E), 256, 0, stream>>>(dst, ew, dinv, E);
  k_dinv  <<<nblk(N), 256, 0, stream>>>(dinv, N);
  k_norm  <<<nblk(E), 256, 0, stream>>>(src, dst, ew, dinv, norm, E);

  // ---- layer 1: H=X@W1 ; agg -> bufA ; x1 = relu(agg + dinv^2*H + b1) in bufA
  k_gemm<<<gemm_blocks, 256, 0, stream>>>(X, W1, H, N);
  hipMemsetAsync(bufA, 0, NC * sizeof(float), stream);
  k_scatter<<<scat_blocks, 256, 0, stream>>>(src, dst, norm, H, bufA, E);
  k_epilogue<<<nblk(NC), 256, 0, stream>>>(bufA, H, dinv, b1, bufA, NC, 1);

  // ---- layer 2: H=x1@W2 ; agg -> bufA ; x2 = relu(...) in bufA
  k_gemm<<<gemm_blocks, 256, 0, stream>>>(bufA, W2, H, N);
  hipMemsetAsync(bufA, 0, NC * sizeof(float), stream);
  k_scatter<<<scat_blocks, 256, 0, stream>>>(src, dst, norm, H, bufA, E);
  k_epilogue<<<nblk(NC), 256, 0, stream>>>(bufA, H, dinv, b2, bufA, NC, 1);

  // ---- layer 3: H=x2@W3 ; agg -> bufA ; out = agg + dinv^2*H + b3 (no relu)
  k_gemm<<<gemm_blocks, 256, 0, stream>>>(bufA, W3, H, N);
  hipMemsetAsync(bufA, 0, NC * sizeof(float), stream);
  k_scatter<<<scat_blocks, 256, 0, stream>>>(src, dst, norm, H, bufA, E);
  k_epilogue<<<nblk(NC), 256, 0, stream>>>(bufA, H, dinv, b3, (float*)d_out, NC, 0);
}